// D_KNN_24567212934029
// MI455X (gfx1250) — compile-verified
//
#include <hip/hip_runtime.h>
#include <hip/hip_bf16.h>
#include <cfloat>

// ---------------------------------------------------------------------------
// Soft-kNN imputation, MI455X (gfx1250, wave32).
// d2 = ||x||^2 - 2 x.q + ||q||^2 ; dot via V_WMMA_F32_16X16X4_F32 (exact fp32).
// ---------------------------------------------------------------------------

typedef float v2f __attribute__((ext_vector_type(2)));
typedef float v8f __attribute__((ext_vector_type(8)));

#define DIM    64
#define NB1    512
#define T1     256
#define HBINS  4096
#define HSCALE 256.0f
#define CAP    4096
#define KSEL   32

__global__ void knn_init(unsigned* __restrict__ ghist, int* __restrict__ candCnt) {
    int t = blockIdx.x * blockDim.x + threadIdx.x;
    if (t < HBINS) ghist[t] = 0u;
    if (t == 0) *candCnt = 0;
}

// Pass 1: distances via WMMA, partial softmax sum, distance histogram.
__global__ void __launch_bounds__(T1)
knn_pass1(const float* __restrict__ X, const float* __restrict__ q,
          float* __restrict__ dist, float* __restrict__ blockZ,
          unsigned* __restrict__ ghist, int N)
{
    __shared__ unsigned hist_s[HBINS];
    __shared__ float red_s[T1];
    for (int i = threadIdx.x; i < HBINS; i += blockDim.x) hist_s[i] = 0u;
    __syncthreads();

    const int lane = threadIdx.x & 31;
    const int hi   = lane >> 4;   // K-half of the 16x4 operand (lanes 16-31 hold K+2,K+3)
    const int m    = lane & 15;   // row within 16-row tile
    const int grp  = lane >> 3;   // 0..3 extraction group

    // B operand: query broadcast across all 16 columns.
    // Assumed 32-bit 4x16 B layout (mirror of documented A 16x4 layout):
    //   VGPR0: lanes0-15 K=k0,   lanes16-31 K=k0+2
    //   VGPR1: lanes0-15 K=k0+1, lanes16-31 K=k0+3
    v2f bq[16];
    const v2f* pq = reinterpret_cast<const v2f*>(q + 2 * hi);
#pragma unroll
    for (int kk = 0; kk < 16; ++kk) bq[kk] = pq[2 * kk];

    float qq = 0.f;
#pragma unroll
    for (int j = 0; j < DIM; ++j) qq = fmaf(q[j], q[j], qq);

    const int wid = (blockIdx.x * blockDim.x + threadIdx.x) >> 5;
    const int nw  = (gridDim.x * blockDim.x) >> 5;
    float zacc = 0.f;

    for (int r0 = wid * 32; r0 < N; r0 += nw * 32) {
        if (r0 + 32 <= N) {                    // wave-uniform branch: EXEC all-ones for WMMA
            const v2f* pA = reinterpret_cast<const v2f*>(X + (size_t)(r0 + m)      * DIM + 2 * hi);
            const v2f* pB = reinterpret_cast<const v2f*>(X + (size_t)(r0 + 16 + m) * DIM + 2 * hi);
            v8f c0 = {}; v8f c1 = {};
            float ps0 = 0.f, ps1 = 0.f;
#pragma unroll
            for (int kk = 0; kk < 16; ++kk) {
                v2f a0 = pA[2 * kk];
                v2f a1 = pB[2 * kk];
                c0 = __builtin_amdgcn_wmma_f32_16x16x4_f32(false, a0, false, bq[kk],
                                                           (short)0, c0, false, false);
                c1 = __builtin_amdgcn_wmma_f32_16x16x4_f32(false, a1, false, bq[kk],
                                                           (short)0, c1, false, false);
                ps0 = fmaf(a0.x, a0.x, fmaf(a0.y, a0.y, ps0));
                ps1 = fmaf(a1.x, a1.x, fmaf(a1.y, a1.y, ps1));
            }
            // combine the two K-halves of the row sum-of-squares (lanes l <-> l^16)
            ps0 += __shfl_xor(ps0, 16, 32);
            ps1 += __shfl_xor(ps1, 16, 32);
            // groups 1,2 need the ssq that lives on lane-8
            int src   = (lane - 8) & 31;
            float sh0 = __shfl(ps0, src, 32);
            float sh1 = __shfl(ps1, src, 32);
            // C/D layout: VGPR j => row j (lanes 0-15) / row 8+j (lanes 16-31); all cols equal.
            int j = lane & 7;
            float dotv, ssq; int row;
            if (grp == 0)      { dotv = c0[j]; ssq = ps0; row = r0 + lane;     }
            else if (grp == 1) { dotv = c1[j]; ssq = sh1; row = r0 + lane + 8; }
            else if (grp == 2) { dotv = c0[j]; ssq = sh0; row = r0 + lane - 8; }
            else               { dotv = c1[j]; ssq = ps1; row = r0 + lane;     }

            float d2 = fmaxf(fmaf(-2.f, dotv, ssq + qq), 0.f);
            float d  = sqrtf(d2);
            dist[row] = d;
            zacc += __expf(-d);                 // TAU = 1
            int b = (int)(d * HSCALE); b = b > (HBINS - 1) ? (HBINS - 1) : b;
            atomicAdd(&hist_s[b], 1u);
        } else {                                // scalar tail (never taken for N=1M)
            int row = r0 + lane;
            if (row < N) {
                const float* xr = X + (size_t)row * DIM;
                float ss = 0.f, dt = 0.f;
                for (int k2 = 0; k2 < DIM; ++k2) {
                    float xv = xr[k2];
                    ss = fmaf(xv, xv, ss);
                    dt = fmaf(xv, q[k2], dt);
                }
                float d = sqrtf(fmaxf(fmaf(-2.f, dt, ss + qq), 0.f));
                dist[row] = d;
                zacc += __expf(-d);
                int b = (int)(d * HSCALE); b = b > (HBINS - 1) ? (HBINS - 1) : b;
                atomicAdd(&hist_s[b], 1u);
            }
        }
    }

    // deterministic block reduction of the partial softmax sum
    red_s[threadIdx.x] = zacc;
    __syncthreads();
    for (int s = T1 / 2; s > 0; s >>= 1) {
        if (threadIdx.x < (unsigned)s) red_s[threadIdx.x] += red_s[threadIdx.x + s];
        __syncthreads();
    }
    if (threadIdx.x == 0) blockZ[blockIdx.x] = red_s[0];

    for (int i = threadIdx.x; i < HBINS; i += blockDim.x) {
        unsigned v = hist_s[i];
        if (v) atomicAdd(&ghist[i], v);
    }
}

// Pass 2: Z = sum(blockZ); histogram scan -> first bin with cum >= KSEL.
__global__ void knn_pass2(const float* __restrict__ blockZ, const unsigned* __restrict__ ghist,
                          float* __restrict__ Zp, int* __restrict__ binStarP)
{
    __shared__ float    zs[256];
    __shared__ unsigned ps[256];
    int t = threadIdx.x;
    zs[t] = blockZ[t] + blockZ[t + 256];   // NB1 == 512
    __syncthreads();
    for (int s = 128; s > 0; s >>= 1) {
        if (t < s) zs[t] += zs[t + s];
        __syncthreads();
    }
    if (t == 0) *Zp = zs[0];

    unsigned h[16]; unsigned csum = 0;
#pragma unroll
    for (int i = 0; i < 16; ++i) { h[i] = ghist[t * 16 + i]; csum += h[i]; }
    ps[t] = csum;
    __syncthreads();
    for (int off = 1; off < 256; off <<= 1) {
        unsigned add = (t >= off) ? ps[t - off] : 0u;
        __syncthreads();
        ps[t] += add;
        __syncthreads();
    }
    unsigned before = (t == 0) ? 0u : ps[t - 1];
    if (before < KSEL && ps[t] >= KSEL) {
        unsigned run = before;
        for (int i = 0; i < 16; ++i) {
            run += h[i];
            if (run >= KSEL) { *binStarP = t * 16 + i; break; }
        }
    }
}

// Pass 3: gather candidates (superset of the exact top-K).
__global__ void knn_pass3(const float* __restrict__ dist, const int* __restrict__ binStarP,
                          float* __restrict__ candD, int* __restrict__ candI,
                          int* __restrict__ candCnt, int N)
{
    int bstar = *binStarP;
    for (int i = blockIdx.x * blockDim.x + threadIdx.x; i < N; i += gridDim.x * blockDim.x) {
        float d = dist[i];
        int b = (int)(d * HSCALE); b = b > (HBINS - 1) ? (HBINS - 1) : b;
        if (b <= bstar) {
            int pos = atomicAdd(candCnt, 1);
            if (pos < CAP) { candD[pos] = d; candI[pos] = i; }
        }
    }
}

// Pass 4: exact top-K by repeated argmin (tie-break on index -> deterministic),
// then out[j] = sum_k exp(-d_k) * y[i_k][j] / Z.
__global__ void knn_pass4(const float* __restrict__ candD, const int* __restrict__ candI,
                          const int* __restrict__ candCnt, const float* __restrict__ Zp,
                          const float* __restrict__ y, float* __restrict__ out)
{
    __shared__ float cd[CAP];
    __shared__ int   ci[CAP];
    __shared__ float bd[256]; __shared__ int bi[256]; __shared__ int bs[256];
    __shared__ float selW[KSEL]; __shared__ int selI[KSEL];
    int t = threadIdx.x;
    int M = *candCnt; if (M > CAP) M = CAP;
    for (int i = t; i < M; i += 256) { cd[i] = candD[i]; ci[i] = candI[i]; }
    if (t < KSEL) { selW[t] = 0.f; selI[t] = 0; }
    __syncthreads();

    int nsel = M < KSEL ? M : KSEL;
    for (int r = 0; r < nsel; ++r) {
        float bestD = FLT_MAX; int bestI = 0x7fffffff; int bestS = 0;
        for (int i = t; i < M; i += 256) {
            float d = cd[i]; int idx = ci[i];
            if (d < bestD || (d == bestD && idx < bestI)) { bestD = d; bestI = idx; bestS = i; }
        }
        bd[t] = bestD; bi[t] = bestI; bs[t] = bestS;
        __syncthreads();
        for (int s = 128; s > 0; s >>= 1) {
            if (t < s) {
                float dB = bd[t + s]; int iB = bi[t + s];
                if (dB < bd[t] || (dB == bd[t] && iB < bi[t])) {
                    bd[t] = dB; bi[t] = iB; bs[t] = bs[t + s];
                }
            }
            __syncthreads();
        }
        if (t == 0) {
            selW[r] = __expf(-bd[0]);
            selI[r] = bi[0];
            cd[bs[0]] = FLT_MAX;      // remove selected candidate
        }
        __syncthreads();
    }

    float Z = *Zp;
    if (t < DIM) {
        float acc = 0.f;
        for (int r = 0; r < KSEL; ++r)
            acc += selW[r] * y[(size_t)selI[r] * DIM + t];
        out[t] = acc / Z;
    }
}

extern "C" void kernel_launch(void* const* d_in, const int* in_sizes, int n_in,
                              void* d_out, int out_size, void* d_ws, size_t ws_size,
                              hipStream_t stream) {
    const float* X = (const float*)d_in[0];   // [N, 64]
    const float* y = (const float*)d_in[1];   // [N, 64]
    const float* q = (const float*)d_in[2];   // [64]
    int N = in_sizes[0] / DIM;

    // workspace layout (~4.05 MB)
    float*    dist    = (float*)d_ws;                       // N
    float*    blockZ  = dist + N;                           // NB1
    unsigned* hist    = (unsigned*)(blockZ + NB1);          // HBINS
    float*    Zp      = (float*)(hist + HBINS);             // 1
    int*      binStar = (int*)(Zp + 1);                     // 1
    int*      candCnt = binStar + 1;                        // 1
    float*    candD   = (float*)(candCnt + 1);              // CAP
    int*      candI   = (int*)(candD + CAP);                // CAP

    knn_init <<<(HBINS + 255) / 256, 256, 0, stream>>>(hist, candCnt);
    knn_pass1<<<NB1, T1, 0, stream>>>(X, q, dist, blockZ, hist, N);
    knn_pass2<<<1, 256, 0, stream>>>(blockZ, hist, Zp, binStar);
    knn_pass3<<<256, 256, 0, stream>>>(dist, binStar, candD, candI, candCnt, N);
    knn_pass4<<<1, 256, 0, stream>>>(candD, candI, candCnt, Zp, y, (float*)d_out);
}